// E2ECR_42451456753706
// MI455X (gfx1250) — compile-verified
//
#include <hip/hip_runtime.h>
#include <hip/hip_bf16.h>
#include <stdint.h>

// CDNA5 / gfx1250 fp32 WMMA types
typedef __attribute__((ext_vector_type(2))) float v2f;
typedef __attribute__((ext_vector_type(8))) float v8f;

#define HID   150
#define HIDP  160   // padded to 10 * 16
#define E     768
#define E3    2304
#define B_    2
#define T_    2048
#define S_    96
#define SPANW 8

// ---------------------------------------------------------------------------
// Weight padding WITH transpose: dst[160 x Kp] <- src[K x 150]^T, zero pad.
// dst[c*Kp + k] = src[k*150 + c]
// ---------------------------------------------------------------------------
__global__ void e2e_pad_mat_t(const float* __restrict__ src, float* __restrict__ dst,
                              int K, int Kp) {
    int total = Kp * HIDP;
    for (int idx = blockIdx.x * blockDim.x + threadIdx.x; idx < total;
         idx += gridDim.x * blockDim.x) {
        int c = idx / Kp, k = idx % Kp;
        dst[idx] = (k < K && c < HID) ? src[k * HID + c] : 0.0f;
    }
}

__global__ void e2e_pad_vec(const float* __restrict__ src, float* __restrict__ dst) {
    int i = threadIdx.x;
    if (i < HIDP) dst[i] = (i < HID) ? src[i] : 0.0f;
}

// ---------------------------------------------------------------------------
// fp32 WMMA GEMM, 4 M-tiles per wave: D[M x 160] = act(A[M x K] @ W + bias)
// W passed TRANSPOSED: WT[160 x ldw], ldw = padded K.
// grid = (M/64, 160/16), block = 32 (one wave). A row-major (lda), D ld=160.
// ---------------------------------------------------------------------------
__global__ __launch_bounds__(32)
void e2e_gemm_f32(const float* __restrict__ A, int lda,
                  const float* __restrict__ WT, int ldw,
                  const float* __restrict__ bias,
                  float* __restrict__ D,
                  int K, int relu_flag) {
    const int mtb  = blockIdx.x * 4;       // first M-tile of this wave
    const int nt   = blockIdx.y;
    const int lane = threadIdx.x;          // 0..31
    const int half = lane >> 4;            // 0: K pair {0,1}; 1: K pair {2,3}
    const int l15  = lane & 15;

    const int bcol = nt * 16 + l15;        // B-matrix: N col per lane
    const float* __restrict__ wr = WT + (size_t)bcol * ldw;
    const float* __restrict__ a0 = A + (size_t)((mtb + 0) * 16 + l15) * lda;
    const float* __restrict__ a1 = A + (size_t)((mtb + 1) * 16 + l15) * lda;
    const float* __restrict__ a2 = A + (size_t)((mtb + 2) * 16 + l15) * lda;
    const float* __restrict__ a3 = A + (size_t)((mtb + 3) * 16 + l15) * lda;

    v8f acc0 = {}, acc1 = {}, acc2 = {}, acc3 = {};
    for (int k0 = 0; k0 < K; k0 += 4) {
        const int kb = k0 + half * 2;
        v2f b, a;
        b.x = wr[kb]; b.y = wr[kb + 1];                    // one b64 load
        a.x = a0[kb]; a.y = a0[kb + 1];
        acc0 = __builtin_amdgcn_wmma_f32_16x16x4_f32(false, a, false, b, (short)0, acc0, false, false);
        a.x = a1[kb]; a.y = a1[kb + 1];
        acc1 = __builtin_amdgcn_wmma_f32_16x16x4_f32(false, a, false, b, (short)0, acc1, false, false);
        a.x = a2[kb]; a.y = a2[kb + 1];
        acc2 = __builtin_amdgcn_wmma_f32_16x16x4_f32(false, a, false, b, (short)0, acc2, false, false);
        a.x = a3[kb]; a.y = a3[kb + 1];
        acc3 = __builtin_amdgcn_wmma_f32_16x16x4_f32(false, a, false, b, (short)0, acc3, false, false);
    }
    const float bv = bias ? bias[bcol] : 0.0f;
#pragma unroll
    for (int mm = 0; mm < 4; ++mm) {
        v8f acc = (mm == 0) ? acc0 : (mm == 1) ? acc1 : (mm == 2) ? acc2 : acc3;
#pragma unroll
        for (int v = 0; v < 8; ++v) {
            const int mrow = (mtb + mm) * 16 + v + half * 8;   // C/D layout
            float x = acc[v] + bv;
            if (relu_flag) x = fmaxf(x, 0.0f);
            D[(size_t)mrow * HIDP + bcol] = x;
        }
    }
}

// ---------------------------------------------------------------------------
// Final 1-wide head: out[r] = H[r,0:150] . w + b  (H has ld 160)
// ---------------------------------------------------------------------------
__global__ void e2e_head_dot(const float* __restrict__ H,
                             const float* __restrict__ w,
                             const float* __restrict__ b,
                             float* __restrict__ out, int M) {
    int r = blockIdx.x * blockDim.x + threadIdx.x;
    if (r >= M) return;
    const float* hr = H + (size_t)r * HIDP;
    float acc = b[0];
    for (int k = 0; k < HID; ++k) acc = fmaf(hr[k], w[k], acc);
    out[r] = acc;
}

// ---------------------------------------------------------------------------
// Span gather: g[b,s] = [X[b,s0], X[b,s0+7], sum_w X[b,s0+w]*attn[b,s0+w]]
// grid = (S, B), block = 256
// ---------------------------------------------------------------------------
__global__ void e2e_gather(const float* __restrict__ X,
                           const int* __restrict__ starts,
                           const float* __restrict__ attn,
                           float* __restrict__ g) {
    const int s = blockIdx.x;
    const int b = blockIdx.y;
    const int s0 = starts[b * S_ + s];
    const float* xb = X + (size_t)b * T_ * E;
    const float* ab = attn + (size_t)b * T_;
    float aw[SPANW];
#pragma unroll
    for (int w = 0; w < SPANW; ++w) aw[w] = ab[s0 + w];
    float* gr = g + ((size_t)b * S_ + s) * E3;
    for (int e = threadIdx.x; e < E; e += blockDim.x) {
        gr[e]         = xb[(size_t)s0 * E + e];
        gr[E + e]     = xb[(size_t)(s0 + SPANW - 1) * E + e];
        float acc = 0.0f;
#pragma unroll
        for (int w = 0; w < SPANW; ++w)
            acc = fmaf(xb[(size_t)(s0 + w) * E + e], aw[w], acc);
        gr[2 * E + e] = acc;
    }
}

// ---------------------------------------------------------------------------
// Fused pairwise kernel.
// grid = (3 row-blocks of 32, 96 j, 2 b); block = 320 (10 waves).
// Wave w owns N-tile w (16 hidden cols) and 2 M-tiles (32 rows).
// gj row staged into LDS via CDNA5 async global->LDS DMA (ASYNCcnt).
// layer1: h1 = relu( (g[i]*g[j]) @ W1c + Apre[i] + Bpre[j] ), sA chunks in LDS
// layer2: h2 = relu(h1 @ W2 + b2);  score = h2 . w3 + b3
// out[b,i,j] = clip((m[i]+m[j]+score)/3, 0, 1)
// ---------------------------------------------------------------------------
#define KC 64
__global__ __launch_bounds__(320)
void e2e_pair(const float* __restrict__ g,
              const float* __restrict__ W1cT,   // [160 x 2304]
              const float* __restrict__ Apre,
              const float* __restrict__ Bpre,
              const float* __restrict__ W2T,    // [160 x 160]
              const float* __restrict__ b2,
              const float* __restrict__ w3,
              const float* __restrict__ b3,
              const float* __restrict__ m,
              float* __restrict__ out) {
    __shared__ __align__(16) float gj[E3];        //  9216 B
    __shared__ __align__(16) float sA[32 * KC];   //  8192 B
    __shared__ __align__(16) float h1[32 * HIDP]; // 20480 B
    __shared__ __align__(16) float h2[32 * HIDP]; // 20480 B

    const int i0   = blockIdx.x * 32;   // first span row of this block
    const int j    = blockIdx.y;        // 0..95
    const int b    = blockIdx.z;        // 0..1
    const int tid  = threadIdx.x;
    const int lane = tid & 31;
    const int wid  = tid >> 5;          // 0..9 -> N-tile
    const int half = lane >> 4;
    const int l15  = lane & 15;
    const int bcol = wid * 16 + l15;    // hidden col (0..159)

    const float* gb  = g + (size_t)b * S_ * E3;
    const float* gbj = gb + (size_t)j * E3;

    // --- async DMA g[b,j,:] (2304 floats = 576 x b128) into LDS ---
    {
        const uint32_t gj_lds = (uint32_t)(uintptr_t)(void*)gj;
        for (int t = tid; t < E3 / 4; t += 320) {
            asm volatile("global_load_async_to_lds_b128 %0, %1, off"
                         :: "v"(gj_lds + (uint32_t)t * 16u),
                            "v"((uint64_t)(uintptr_t)(gbj + t * 4))
                         : "memory");
        }
        asm volatile("s_wait_asynccnt 0x0" ::: "memory");
    }
    __syncthreads();

    // --- layer 1: (g_i * g_j) @ W1c, K = 2304, staged in KC-chunks ---
    const float* __restrict__ w1r = W1cT + (size_t)bcol * E3;
    v8f acc0 = {}, acc1 = {};
    for (int kc = 0; kc < E3; kc += KC) {
        __syncthreads();   // previous chunk fully consumed
        for (int idx = tid; idx < 32 * KC; idx += 320) {
            const int r = idx >> 6, kk = idx & (KC - 1);
            sA[idx] = gb[(size_t)(i0 + r) * E3 + kc + kk] * gj[kc + kk];
        }
        __syncthreads();
        for (int k0 = 0; k0 < KC; k0 += 4) {
            const int kb = k0 + half * 2;
            v2f bv, a;
            bv.x = w1r[kc + kb]; bv.y = w1r[kc + kb + 1];  // one b64 load
            a.x = sA[l15 * KC + kb];        a.y = sA[l15 * KC + kb + 1];
            acc0 = __builtin_amdgcn_wmma_f32_16x16x4_f32(false, a, false, bv, (short)0, acc0, false, false);
            a.x = sA[(16 + l15) * KC + kb]; a.y = sA[(16 + l15) * KC + kb + 1];
            acc1 = __builtin_amdgcn_wmma_f32_16x16x4_f32(false, a, false, bv, (short)0, acc1, false, false);
        }
    }
    const float* AP  = Apre + ((size_t)b * S_ + i0) * HIDP;     // includes b1
    const float  bpj = Bpre[((size_t)b * S_ + j) * HIDP + bcol];
#pragma unroll
    for (int v = 0; v < 8; ++v) {
        int il = v + half * 8;
        h1[il * HIDP + bcol] = fmaxf(acc0[v] + AP[(size_t)il * HIDP + bcol] + bpj, 0.0f);
        il += 16;
        h1[il * HIDP + bcol] = fmaxf(acc1[v] + AP[(size_t)il * HIDP + bcol] + bpj, 0.0f);
    }
    __syncthreads();

    // --- layer 2: h1 (LDS) @ W2, K = 160 ---
    const float* __restrict__ w2r = W2T + (size_t)bcol * HIDP;
    v8f acc2 = {}, acc3 = {};
    for (int k0 = 0; k0 < HIDP; k0 += 4) {
        const int kb = k0 + half * 2;
        v2f bv, a;
        bv.x = w2r[kb]; bv.y = w2r[kb + 1];
        a.x = h1[l15 * HIDP + kb];        a.y = h1[l15 * HIDP + kb + 1];
        acc2 = __builtin_amdgcn_wmma_f32_16x16x4_f32(false, a, false, bv, (short)0, acc2, false, false);
        a.x = h1[(16 + l15) * HIDP + kb]; a.y = h1[(16 + l15) * HIDP + kb + 1];
        acc3 = __builtin_amdgcn_wmma_f32_16x16x4_f32(false, a, false, bv, (short)0, acc3, false, false);
    }
    const float b2v = b2[bcol];
#pragma unroll
    for (int v = 0; v < 8; ++v) {
        int il = v + half * 8;
        h2[il * HIDP + bcol] = fmaxf(acc2[v] + b2v, 0.0f);
        il += 16;
        h2[il * HIDP + bcol] = fmaxf(acc3[v] + b2v, 0.0f);
    }
    __syncthreads();

    // --- head + combine + clip ---
    if (tid < 32) {
        const int i = i0 + tid;
        float s = b3[0];
        for (int k = 0; k < HID; ++k) s = fmaf(h2[tid * HIDP + k], w3[k], s);
        const float mi = m[b * S_ + i];
        const float mj = m[b * S_ + j];
        float e = (mi + mj + s) / 3.0f;
        e = fminf(fmaxf(e, 0.0f), 1.0f);
        out[((size_t)b * S_ + i) * S_ + j] = e;
    }
}

// ---------------------------------------------------------------------------
// Host launcher
// ---------------------------------------------------------------------------
extern "C" void kernel_launch(void* const* d_in, const int* in_sizes, int n_in,
                              void* d_out, int out_size, void* d_ws, size_t ws_size,
                              hipStream_t stream) {
    const float* X      = (const float*)d_in[0];   // [2,2048,768]
    const int*   starts = (const int*)  d_in[1];   // [2,96]
    const float* aw1 = (const float*)d_in[2],  *ab1 = (const float*)d_in[3];
    const float* aw2 = (const float*)d_in[4],  *ab2 = (const float*)d_in[5];
    const float* aw3 = (const float*)d_in[6],  *ab3 = (const float*)d_in[7];
    const float* mw1 = (const float*)d_in[8],  *mb1 = (const float*)d_in[9];
    const float* mw2 = (const float*)d_in[10], *mb2 = (const float*)d_in[11];
    const float* mw3 = (const float*)d_in[12], *mb3 = (const float*)d_in[13];
    const float* pw1 = (const float*)d_in[14], *pb1 = (const float*)d_in[15];
    const float* pw2 = (const float*)d_in[16], *pb2 = (const float*)d_in[17];
    const float* pw3 = (const float*)d_in[18], *pb3 = (const float*)d_in[19];
    float* out = (float*)d_out;

    // workspace carve-up (floats); total ~14.2 MB
    float* w = (float*)d_ws;
    size_t o = 0;
    auto alloc = [&](size_t n) { float* p = w + o; o += n; return p; };
    float* aw1T = alloc((size_t)HIDP * E);     // transposed padded weights
    float* ab1p = alloc(HIDP);
    float* aw2T = alloc((size_t)HIDP * HIDP);
    float* ab2p = alloc(HIDP);
    float* mw1T = alloc((size_t)HIDP * E3);
    float* mb1p = alloc(HIDP);
    float* mw2T = alloc((size_t)HIDP * HIDP);
    float* mb2p = alloc(HIDP);
    float* W1aT = alloc((size_t)HIDP * E3);
    float* W1bT = alloc((size_t)HIDP * E3);
    float* W1cT = alloc((size_t)HIDP * E3);
    float* pb1p = alloc(HIDP);
    float* pw2T = alloc((size_t)HIDP * HIDP);
    float* pb2p = alloc(HIDP);
    float* H1   = alloc((size_t)B_ * T_ * HIDP);
    float* H2   = alloc((size_t)B_ * T_ * HIDP);
    float* attn = alloc((size_t)B_ * T_);
    float* gbuf = alloc((size_t)B_ * S_ * E3);
    float* gH1  = alloc((size_t)B_ * S_ * HIDP);
    float* gH2  = alloc((size_t)B_ * S_ * HIDP);
    float* mS   = alloc((size_t)B_ * S_);
    float* Apre = alloc((size_t)B_ * S_ * HIDP);
    float* Bpre = alloc((size_t)B_ * S_ * HIDP);
    (void)ws_size; (void)n_in; (void)in_sizes; (void)out_size;

    auto padg = [](int Kp) { return dim3((unsigned)((Kp * HIDP + 255) / 256)); };

    // weight padding + transpose
    e2e_pad_mat_t<<<padg(E),   256, 0, stream>>>(aw1, aw1T, E,   E);
    e2e_pad_mat_t<<<padg(HIDP),256, 0, stream>>>(aw2, aw2T, HID, HIDP);
    e2e_pad_mat_t<<<padg(E3),  256, 0, stream>>>(mw1, mw1T, E3,  E3);
    e2e_pad_mat_t<<<padg(HIDP),256, 0, stream>>>(mw2, mw2T, HID, HIDP);
    e2e_pad_mat_t<<<padg(E3),  256, 0, stream>>>(pw1,                     W1aT, E3, E3);
    e2e_pad_mat_t<<<padg(E3),  256, 0, stream>>>(pw1 + (size_t)E3 * HID,     W1bT, E3, E3);
    e2e_pad_mat_t<<<padg(E3),  256, 0, stream>>>(pw1 + (size_t)2 * E3 * HID, W1cT, E3, E3);
    e2e_pad_mat_t<<<padg(HIDP),256, 0, stream>>>(pw2, pw2T, HID, HIDP);
    e2e_pad_vec<<<1, HIDP, 0, stream>>>(ab1, ab1p);
    e2e_pad_vec<<<1, HIDP, 0, stream>>>(ab2, ab2p);
    e2e_pad_vec<<<1, HIDP, 0, stream>>>(mb1, mb1p);
    e2e_pad_vec<<<1, HIDP, 0, stream>>>(mb2, mb2p);
    e2e_pad_vec<<<1, HIDP, 0, stream>>>(pb1, pb1p);
    e2e_pad_vec<<<1, HIDP, 0, stream>>>(pb2, pb2p);

    // attention MLP over all B*T tokens (M = 4096)
    e2e_gemm_f32<<<dim3(B_ * T_ / 64, HIDP / 16), 32, 0, stream>>>(X, E,    aw1T, E,    ab1p, H1, E,    1);
    e2e_gemm_f32<<<dim3(B_ * T_ / 64, HIDP / 16), 32, 0, stream>>>(H1, HIDP, aw2T, HIDP, ab2p, H2, HIDP, 1);
    e2e_head_dot<<<(B_ * T_) / 256, 256, 0, stream>>>(H2, aw3, ab3, attn, B_ * T_);

    // span features g = [g_start, g_end, g_attn]
    e2e_gather<<<dim3(S_, B_), 256, 0, stream>>>(X, starts, attn, gbuf);

    // mention scores (M = 192)
    e2e_gemm_f32<<<dim3(B_ * S_ / 64, HIDP / 16), 32, 0, stream>>>(gbuf, E3,  mw1T, E3,   mb1p, gH1, E3,   1);
    e2e_gemm_f32<<<dim3(B_ * S_ / 64, HIDP / 16), 32, 0, stream>>>(gH1, HIDP, mw2T, HIDP, mb2p, gH2, HIDP, 1);
    e2e_head_dot<<<1, 256, 0, stream>>>(gH2, mw3, mb3, mS, B_ * S_);

    // pairwise layer-1 per-span precomputes (bias folded into Apre)
    e2e_gemm_f32<<<dim3(B_ * S_ / 64, HIDP / 16), 32, 0, stream>>>(gbuf, E3, W1aT, E3, pb1p,    Apre, E3, 0);
    e2e_gemm_f32<<<dim3(B_ * S_ / 64, HIDP / 16), 32, 0, stream>>>(gbuf, E3, W1bT, E3, nullptr, Bpre, E3, 0);

    // fused pairwise scoring
    e2e_pair<<<dim3(S_ / 32, S_, B_), 320, 0, stream>>>(
        gbuf, W1cT, Apre, Bpre, pw2T, pb2p, pw3, pb3, mS, out);
}